// GraphConvolution_cu_21560735826551
// MI455X (gfx1250) — compile-verified
//
#include <hip/hip_runtime.h>
#include <hip/hip_bf16.h>

typedef float v2f __attribute__((ext_vector_type(2)));
typedef float v8f __attribute__((ext_vector_type(8)));

#define IN_F   128
#define OUT_F  128
#define NNODES 100000
#define LDS_PAD 4          // row stride 132 floats -> bank stride 4, conflict-free A reads

// ---------------------------------------------------------------------------
// Kernel 1: support = X @ W  via V_WMMA_F32_16X16X4_F32 (full fp32 precision).
// Block = 256 threads = 8 waves; block owns one 16-row M tile, wave w owns
// N tile w (16 cols). The 16x128 X tile is staged once in LDS (8 KB) so the
// 8 waves don't re-fetch it 8x from global (saves ~360 MB of HBM/L2 reads).
// ---------------------------------------------------------------------------
__global__ __launch_bounds__(256)
void gcn_gemm_wmma(const float* __restrict__ X, const float* __restrict__ W,
                   float* __restrict__ S, int nrows)
{
    __shared__ float As[16][IN_F + LDS_PAD];

    const int tid  = threadIdx.x;
    const int lane = tid & 31;
    const int wave = tid >> 5;                   // 0..7 -> N tile
    const int m0   = blockIdx.x * 16;            // grid sized exactly: m0 < nrows

    // ---- cooperative stage: 16x128 floats = 512 float4, 2 per thread ----
    {
        int idx = tid;                           // float4 index
#pragma unroll
        for (int r = 0; r < 2; ++r, idx += 256) {
            const int row  = idx >> 5;           // 32 float4 per row
            const int col4 = idx & 31;
            float4 v = *(const float4*)(X + (size_t)(m0 + row) * IN_F + col4 * 4);
            *(float4*)&As[row][col4 * 4] = v;    // ds_store_b128
        }
    }
    __syncthreads();

    const int half = lane >> 4;                  // K sub-pair select per ISA layout
    const int l15  = lane & 15;
    const int n0   = wave * 16;

    v8f c = {0.f, 0.f, 0.f, 0.f, 0.f, 0.f, 0.f, 0.f};

#pragma unroll 8
    for (int k = 0; k < IN_F; k += 4) {
        const int ka = k + 2 * half;
        v2f a = *(const v2f*)&As[l15][ka];              // ds_load_b64, bank-conflict-free
        v2f b;
        b.x = W[(size_t)ka       * OUT_F + n0 + l15];   // B[ka][n]   (hot in WGP$/L2)
        b.y = W[(size_t)(ka + 1) * OUT_F + n0 + l15];   // B[ka+1][n]
        // 8 args: (neg_a, A, neg_b, B, c_mod, C, reuse_a, reuse_b)
        c = __builtin_amdgcn_wmma_f32_16x16x4_f32(false, a, false, b,
                                                  (short)0, c, false, false);
    }

    // C/D layout: VGPR r holds row M = r + 8*half, col = n0 + l15
    float* srow = S + (size_t)(m0 + 8 * half) * OUT_F + n0 + l15;
#pragma unroll
    for (int r = 0; r < 8; ++r)
        srow[(size_t)r * OUT_F] = c[r];
    (void)nrows;
}

// ---------------------------------------------------------------------------
// Kernel 0: zero the accumulator + degree counters in workspace.
// ---------------------------------------------------------------------------
__global__ __launch_bounds__(256)
void gcn_zero(float4* __restrict__ agg4, int* __restrict__ cnt,
              int n_agg4, int n_cnt)
{
    int i = blockIdx.x * blockDim.x + threadIdx.x;
    if (i < n_agg4) agg4[i] = make_float4(0.f, 0.f, 0.f, 0.f);
    if (i < n_cnt)  cnt[i]  = 0;
}

// ---------------------------------------------------------------------------
// Kernel 2: edge scatter. One wave per edge: gather support[src] (float4 per
// lane = 512B coalesced per wave, L2-resident) and atomic-add into agg[dst]
// (f32 atomics execute at L2). Lane 0 bumps the in-degree counter.
// ---------------------------------------------------------------------------
__global__ __launch_bounds__(256)
void gcn_scatter(const int* __restrict__ src, const int* __restrict__ dst,
                 const float* __restrict__ S, float* __restrict__ agg,
                 int* __restrict__ cnt, int nedges)
{
    const int tid  = blockIdx.x * blockDim.x + threadIdx.x;
    const int e    = tid >> 5;
    const int lane = tid & 31;
    if (e >= nedges) return;

    const int s = src[e];
    const int d = dst[e];

    const float4 v = *(const float4*)(S + (size_t)s * OUT_F + lane * 4);
    float* ap = agg + (size_t)d * OUT_F + lane * 4;
    atomicAdd(ap + 0, v.x);
    atomicAdd(ap + 1, v.y);
    atomicAdd(ap + 2, v.z);
    atomicAdd(ap + 3, v.w);

    if (lane == 0) atomicAdd(&cnt[d], 1);
}

// ---------------------------------------------------------------------------
// Kernel 3: out[i] = (agg[i] + support[i]) / (cnt[i] + 1) + b
// Self-loop folded in analytically (saves 100K waves of atomics).
// ---------------------------------------------------------------------------
__global__ __launch_bounds__(256)
void gcn_finalize(const float* __restrict__ agg, const float* __restrict__ S,
                  const int* __restrict__ cnt, const float* __restrict__ bias,
                  float* __restrict__ out, int nnodes)
{
    const int i = blockIdx.x * blockDim.x + threadIdx.x;   // one float4 per thread
    const int node = i >> 5;                               // 128/4 = 32 groups/node
    const int g    = i & 31;
    if (node >= nnodes) return;

    const size_t off = (size_t)node * OUT_F + g * 4;
    const float4 a = *(const float4*)(agg + off);
    const float4 s = *(const float4*)(S + off);
    const float4 b = *(const float4*)(bias + g * 4);
    const float inv = 1.0f / (float)(cnt[node] + 1);

    float4 o;
    o.x = (a.x + s.x) * inv + b.x;
    o.y = (a.y + s.y) * inv + b.y;
    o.z = (a.z + s.z) * inv + b.z;
    o.w = (a.w + s.w) * inv + b.w;
    *(float4*)(out + off) = o;
}

// ---------------------------------------------------------------------------
extern "C" void kernel_launch(void* const* d_in, const int* in_sizes, int n_in,
                              void* d_out, int out_size, void* d_ws, size_t ws_size,
                              hipStream_t stream)
{
    const float* x      = (const float*)d_in[0];   // [N, 128]
    const int*   src    = (const int*)d_in[1];     // [E]
    const int*   dst    = (const int*)d_in[2];     // [E]
    const float* weight = (const float*)d_in[3];   // [128, 128]
    const float* bias   = (const float*)d_in[4];   // [128]
    float*       out    = (float*)d_out;           // [N, 128]

    const int E = in_sizes[1];

    // Workspace layout: support | agg | cnt
    float* S   = (float*)d_ws;                                   // 51.2 MB
    float* agg = S + (size_t)NNODES * OUT_F;                     // 51.2 MB
    int*   cnt = (int*)(agg + (size_t)NNODES * OUT_F);           // 0.4 MB

    const int n_agg4 = NNODES * OUT_F / 4;                       // 3.2M float4s

    gcn_zero<<<(n_agg4 + 255) / 256, 256, 0, stream>>>(
        (float4*)agg, cnt, n_agg4, NNODES);

    gcn_gemm_wmma<<<NNODES / 16, 256, 0, stream>>>(x, weight, S, NNODES);

    const long long nthreads_sc = (long long)E * 32;
    gcn_scatter<<<(unsigned)((nthreads_sc + 255) / 256), 256, 0, stream>>>(
        src, dst, S, agg, cnt, E);

    const int nthreads_fin = NNODES * 32;
    gcn_finalize<<<(nthreads_fin + 255) / 256, 256, 0, stream>>>(
        agg, S, cnt, bias, out, NNODES);
}